// AdaptiveNodeClassifier_42339787604745
// MI455X (gfx1250) — compile-verified
//
#include <hip/hip_runtime.h>
#include <math.h>

typedef float v2f __attribute__((ext_vector_type(2)));
typedef float v8f __attribute__((ext_vector_type(8)));
typedef unsigned int v4u __attribute__((ext_vector_type(4)));
typedef int v4i __attribute__((ext_vector_type(4)));
typedef int v8i __attribute__((ext_vector_type(8)));

#define HID 128
#define NCLS 10
#define MIX 0.5f       /* T_HET * HP_MAX */
#define INJ_LAM 0.8f

// ---------------------------------------------------------------- init / norm
__global__ void init_kernel(float* __restrict__ deg, int* __restrict__ flags, int n) {
  int i = blockIdx.x * blockDim.x + threadIdx.x;
  if (i < n) { deg[i] = 1.0f; flags[i] = 0; }   // self-loop contributes 1 to every node
}

__global__ void deg_kernel(const int* __restrict__ dst, float* __restrict__ deg, int e) {
  int i = blockIdx.x * blockDim.x + threadIdx.x;
  if (i < e) atomicAdd(&deg[dst[i]], 1.0f);
}

__global__ void rsqrt_kernel(float* __restrict__ d, int n) {
  int i = blockIdx.x * blockDim.x + threadIdx.x;
  if (i < n) d[i] = rsqrtf(d[i]);
}

// ---------------------------------------------------------------- TDM staging
// Stage the full 128x128 f32 weight matrix (64 KB) into LDS with one Tensor
// Data Mover descriptor: 2-D tensor, data_size=4B, tile == tensor.
// D# group0: count=1 | lds_addr | global_addr | type=2.  group1: data_size=2,
// tensor_dim0/1 = 128, tile_dim0/1 = 128, tile_dim2 = 0, dim0_stride = 128.
__device__ __forceinline__ void tdm_load_w128(const float* W, float* sW) {
  unsigned long long ga = (unsigned long long)(uintptr_t)W;
  unsigned int lds = (unsigned int)(uintptr_t)sW;  // flat-LDS addr[31:0] == LDS byte offset
  v4u g0;
  g0.x = 1u;                                         // count=1, user descriptor
  g0.y = lds;                                        // lds_addr [63:32]
  g0.z = (unsigned int)(ga & 0xFFFFFFFFu);           // global_addr [95:64]
  g0.w = (unsigned int)((ga >> 32) & 0x1FFFFFFu)     // global_addr [120:96]
       | (2u << 30);                                 // type=2 ("image") [127:126]
  v8i g1;
  g1[0] = 0x00020000;                                // wg_mask=0, data_size=2 (4B)
  g1[1] = (int)(128u << 16);                         // tensor_dim0 = 128 (bits 79:48)
  g1[2] = (int)(128u << 16);                         // tensor_dim1 = 128 (bits 111:80)
  g1[3] = (int)(128u << 16);                         // tile_dim0   = 128 (bits 127:112)
  g1[4] = 128;                                       // tile_dim1 = 128, tile_dim2 = 0
  g1[5] = 128;                                       // tensor_dim0_stride = 128 (lo 32)
  g1[6] = 0;                                         // stride0 hi / stride1 lo
  g1[7] = 0;
  v4i z4 = {0, 0, 0, 0};                             // groups 2/3 unused (2-D tensor)
  v8i z8 = {0, 0, 0, 0, 0, 0, 0, 0};
  __builtin_amdgcn_tensor_load_to_lds(g0, g1, z4, z4, z8, 0);
}

// ---------------------------------------------------------------- WMMA GEMM
// Y[nrows x 128] = X[nrows x 128] @ W[128 x 128] with fused epilogues.
// One block = 256 threads = 8 waves; each wave computes a 16x128 strip with
// v_wmma_f32_16x16x4_f32 (fp32-exact; GEMM FLOPs are negligible vs BW here).
// EPI: 1 = relu(val + bias)                              (head MLP)
//      2 = Y=val; Y2=val*dinv[row]^2 + bias              (conv self-loop init)
//      3 = Y = low + MIX*(val - low)                     (hetero mix)
//      4 = relu of mode 3
template <int EPI>
__global__ __launch_bounds__(256) void gemm128_wmma(const float* __restrict__ X,
                                                    const float* __restrict__ W,
                                                    const float* __restrict__ bias,
                                                    const float* __restrict__ dinv,
                                                    const float* __restrict__ lowbuf,
                                                    float* __restrict__ Y,
                                                    float* __restrict__ Y2, int nrows) {
  __shared__ float sW[HID * HID];
  if (threadIdx.x < 32) {                 // wave 0 issues the TDM DMA
    tdm_load_w128(W, sW);
    __builtin_amdgcn_s_wait_tensorcnt(0);
  }
  __syncthreads();

  const int wave = threadIdx.x >> 5;
  const int lane = threadIdx.x & 31;
  const int row0 = blockIdx.x * 128 + wave * 16;
  if (row0 >= nrows) return;              // nrows % 16 == 0: whole 16-row tiles only

  const int hrow = lane & 15;             // A-matrix M within tile
  const int koff = (lane >> 4) * 2;       // A/B K sub-offset per half-wave
  const int ncol = lane & 15;             // B/C/D N within tile

  v8f acc[8];
#pragma unroll
  for (int t = 0; t < 8; ++t) acc[t] = 0.0f;

  const float* xrow = X + (long long)(row0 + hrow) * HID;
  for (int kk = 0; kk < HID; kk += 4) {
    const float2 av = *(const float2*)(xrow + kk + koff);
    v2f a; a.x = av.x; a.y = av.y;
#pragma unroll
    for (int t = 0; t < 8; ++t) {
      v2f b;
      b.x = sW[(kk + koff) * HID + t * 16 + ncol];
      b.y = sW[(kk + koff + 1) * HID + t * 16 + ncol];
      acc[t] = __builtin_amdgcn_wmma_f32_16x16x4_f32(false, a, false, b,
                                                     (short)0, acc[t], false, false);
    }
  }

  [[maybe_unused]] float di[8];
  if constexpr (EPI == 2) {
#pragma unroll
    for (int r = 0; r < 8; ++r) di[r] = dinv[row0 + r + (lane >> 4) * 8];
  }

  // C/D layout: VGPR r -> (M=r, N=lane) lanes 0-15; (M=r+8, N=lane-16) lanes 16-31
#pragma unroll
  for (int t = 0; t < 8; ++t) {
#pragma unroll
    for (int r = 0; r < 8; ++r) {
      const int orow = row0 + r + (lane >> 4) * 8;
      const int ocol = t * 16 + ncol;
      const long long idx = (long long)orow * HID + ocol;
      const float v = acc[t][r];
      if constexpr (EPI == 1) {
        Y[idx] = fmaxf(v + bias[ocol], 0.0f);
      } else if constexpr (EPI == 2) {
        Y[idx] = v;
        Y2[idx] = v * di[r] * di[r] + bias[ocol];
      } else if constexpr (EPI == 3) {
        const float lo = lowbuf[idx];
        Y[idx] = lo + MIX * (v - lo);
      } else if constexpr (EPI == 4) {
        const float lo = lowbuf[idx];
        Y[idx] = fmaxf(lo + MIX * (v - lo), 0.0f);
      } else {
        Y[idx] = v;
      }
    }
  }
}

// ---------------------------------------------------------------- GCN scatter
// out[dst] += dinv[src]*dinv[dst] * hw[src]   (one thread per edge-feature)
__global__ void scatter_kernel(const int* __restrict__ src, const int* __restrict__ dst,
                               const float* __restrict__ dinv, const float* __restrict__ hw,
                               float* __restrict__ out, int e) {
  int idx = blockIdx.x * blockDim.x + threadIdx.x;
  if (idx >= e * HID) return;
  const int ei = idx >> 7, f = idx & (HID - 1);
  const int s = src[ei], d = dst[ei];
  const float w = dinv[s] * dinv[d];
  atomicAdd(&out[(long long)d * HID + f], hw[(long long)s * HID + f] * w);
}

// ---------------------------------------------------------------- injection + head
__global__ void flag_kernel(const int* __restrict__ inj, int* __restrict__ flags, int n) {
  int i = blockIdx.x * blockDim.x + threadIdx.x;
  if (i < n) flags[inj[i]] = 1;    // duplicate stores of identical value: race-free
}

// One wave per node: logits = m . Wm2 + bm2 (shuffle reduction), set-style
// injection add (once per unique node via flag), log_softmax, write 10 outputs.
__global__ __launch_bounds__(256) void head_kernel(const float* __restrict__ m,
                                                   const float* __restrict__ Wm2,
                                                   const float* __restrict__ bm2,
                                                   const int* __restrict__ flags,
                                                   const float* __restrict__ preds,
                                                   float* __restrict__ out, int n) {
  const int node = blockIdx.x * 8 + (threadIdx.x >> 5);
  const int lane = threadIdx.x & 31;
  if (node >= n) return;

  const float4 mv = *(const float4*)(m + (long long)node * HID + lane * 4);
  float lg[NCLS];
#pragma unroll
  for (int c = 0; c < NCLS; ++c) {
    const float* wc = Wm2 + c;
    float p = mv.x * wc[(lane * 4 + 0) * NCLS] + mv.y * wc[(lane * 4 + 1) * NCLS] +
              mv.z * wc[(lane * 4 + 2) * NCLS] + mv.w * wc[(lane * 4 + 3) * NCLS];
#pragma unroll
    for (int off = 16; off > 0; off >>= 1) p += __shfl_xor(p, off, 32);
    lg[c] = p + bm2[c];
  }

  if (flags[node]) {
#pragma unroll
    for (int c = 0; c < NCLS; ++c) lg[c] += INJ_LAM * preds[(long long)node * NCLS + c];
  }

  float mx = lg[0];
#pragma unroll
  for (int c = 1; c < NCLS; ++c) mx = fmaxf(mx, lg[c]);
  float s = 0.0f;
#pragma unroll
  for (int c = 0; c < NCLS; ++c) s += expf(lg[c] - mx);
  const float lse = mx + logf(s);
  if (lane < NCLS) out[(long long)node * NCLS + lane] = lg[lane] - lse;
}

// ---------------------------------------------------------------- launch
extern "C" void kernel_launch(void* const* d_in, const int* in_sizes, int n_in,
                              void* d_out, int out_size, void* d_ws, size_t ws_size,
                              hipStream_t stream) {
  (void)n_in; (void)out_size; (void)ws_size;
  const float* x     = (const float*)d_in[0];
  const int*   ei    = (const int*)d_in[1];
  const int*   inj   = (const int*)d_in[2];
  const float* preds = (const float*)d_in[3];
  const float* W1  = (const float*)d_in[4];
  const float* b1  = (const float*)d_in[5];
  const float* W2  = (const float*)d_in[6];
  const float* b2  = (const float*)d_in[7];
  const float* Wl1 = (const float*)d_in[8];
  const float* Wl2 = (const float*)d_in[9];
  const float* Wm1 = (const float*)d_in[10];
  const float* bm1 = (const float*)d_in[11];
  const float* Wm2 = (const float*)d_in[12];
  const float* bm2 = (const float*)d_in[13];

  const int n    = in_sizes[0] / HID;   // 100000
  const int e    = in_sizes[1] / 2;     // 1600000
  const int ninj = in_sizes[2];         // 20000
  const int* src = ei;
  const int* dst = ei + e;

  float* A    = (float*)d_ws;                 // N*128  (hw / m)
  float* B    = A + (size_t)n * HID;          // N*128  (low)
  float* Cb   = B + (size_t)n * HID;          // N*128  (h)
  float* Db   = Cb + (size_t)n * HID;         // N*128  (h2)
  float* dinv = Db + (size_t)n * HID;         // N
  int*  flags = (int*)(dinv + n);             // N

  const int gb = (n + 127) / 128;             // GEMM grid
  const int eb = (int)(((long long)e * HID + 255) / 256);
  const dim3 T(256);

  // normalization
  init_kernel<<<(n + 255) / 256, T, 0, stream>>>(dinv, flags, n);
  deg_kernel<<<(e + 255) / 256, T, 0, stream>>>(dst, dinv, e);
  rsqrt_kernel<<<(n + 255) / 256, T, 0, stream>>>(dinv, n);

  // layer 1:  A = x@W1,  B = A*dinv^2 + b1  (fused), then edge scatter,
  //           Cb = relu(B + MIX*(x@Wl1 - B))  (mix fused into GEMM epilogue)
  gemm128_wmma<2><<<gb, T, 0, stream>>>(x, W1, b1, dinv, nullptr, A, B, n);
  scatter_kernel<<<eb, T, 0, stream>>>(src, dst, dinv, A, B, e);
  gemm128_wmma<4><<<gb, T, 0, stream>>>(x, Wl1, nullptr, nullptr, B, Cb, nullptr, n);

  // layer 2
  gemm128_wmma<2><<<gb, T, 0, stream>>>(Cb, W2, b2, dinv, nullptr, A, B, n);
  scatter_kernel<<<eb, T, 0, stream>>>(src, dst, dinv, A, B, e);
  gemm128_wmma<3><<<gb, T, 0, stream>>>(Cb, Wl2, nullptr, nullptr, B, Db, nullptr, n);

  // head: A = relu(h2@Wm1 + bm1); flags; fused classifier + injection + log_softmax
  gemm128_wmma<1><<<gb, T, 0, stream>>>(Db, Wm1, bm1, nullptr, nullptr, A, nullptr, n);
  flag_kernel<<<(ninj + 255) / 256, T, 0, stream>>>(inj, flags, ninj);
  head_kernel<<<(n + 7) / 8, T, 0, stream>>>(A, Wm2, bm2, flags, preds, (float*)d_out, n);
}